// GATHead_65154653880995
// MI455X (gfx1250) — compile-verified
//
#include <hip/hip_runtime.h>
#include <math.h>

// ---------------------------------------------------------------------------
// GAT head for MI455X (gfx1250, wave32). Both GEMMs use V_WMMA_F32_16X16X4_F32
// so the matrix pipe is exercised at full fp32 fidelity to the f32 reference.
// ---------------------------------------------------------------------------

typedef __attribute__((ext_vector_type(2))) float v2f;
typedef __attribute__((ext_vector_type(8))) float v8f;

#define WMMA_F32_4(A, B, C) \
  __builtin_amdgcn_wmma_f32_16x16x4_f32(false, (A), false, (B), (short)0, (C), false, false)

static __device__ __forceinline__ float lrelu02(float x) {
  return x > 0.f ? x : 0.2f * x;
}

// WMMA f32 16x16x4 per-lane fragment coordinates (ISA 7.12.2):
//   A (16x4, MxK):  M = lane%16,              K = 2*(lane/16) + vgpr{0,1}
//   B (4x16, KxN):  N = lane%16,              K = 2*(lane/16) + vgpr{0,1}
//   C/D (16x16):    M = vgpr + 8*(lane/16),   N = lane%16

// ---------------------------------------------------------------------------
// Kernel 1: H[b,i,f] = sum_k X[b,i,k] * W[k,f]   ([2,1024,128]@[128,64])
// grid = B*(N/16) = 128 blocks, 128 threads = 4 waves; wave t owns cols [16t,16t+16)
// ---------------------------------------------------------------------------
__global__ void k_gemm_h(const float* __restrict__ X, const float* __restrict__ W,
                         float* __restrict__ H) {
  const int b    = blockIdx.x >> 6;          // 64 row-tiles per batch
  const int i0   = (blockIdx.x & 63) << 4;   // row-tile base
  const int t    = threadIdx.x >> 5;         // n-tile 0..3
  const int lane = threadIdx.x & 31;
  const int mi = lane & 15, hh = lane >> 4;

  const float* xrow = X + ((size_t)(b * 1024 + i0 + mi)) * 128;
  v8f acc = {};
  #pragma unroll 4
  for (int kk = 0; kk < 128; kk += 4) {
    const int kr = kk + 2 * hh;
    v2f a;  a[0] = xrow[kr];                    a[1] = xrow[kr + 1];
    v2f bb; bb[0] = W[kr * 64 + t * 16 + mi];   bb[1] = W[(kr + 1) * 64 + t * 16 + mi];
    acc = WMMA_F32_4(a, bb, acc);
  }
  #pragma unroll
  for (int r = 0; r < 8; ++r) {
    const int row = i0 + r + 8 * hh;
    H[((size_t)(b * 1024 + row)) * 64 + t * 16 + mi] = acc[r];
  }
}

// ---------------------------------------------------------------------------
// Kernel 2: u[r]=a1.h[r], v[r]=a2.h[r];
//   expA[b,r]  = exp(lrelu(u[r]+v[r]))                (rows i<512 use r=2i+(j>=512))
//   expB[b,jj] = exp(lrelu(u[2jj]+v[2jj+1]))          (rows i>=512, per column)
// grid = B blocks, 1024 threads
// ---------------------------------------------------------------------------
__global__ void k_scores(const float* __restrict__ H, const float* __restrict__ avec,
                         float* __restrict__ expA, float* __restrict__ expB) {
  __shared__ float su[1024];
  __shared__ float sv[1024];
  const int b = blockIdx.x, r = threadIdx.x;
  const float* hr = H + ((size_t)(b * 1024 + r)) * 64;
  float u = 0.f, v = 0.f;
  #pragma unroll 8
  for (int f = 0; f < 64; ++f) {
    const float hv = hr[f];
    u = fmaf(avec[f], hv, u);
    v = fmaf(avec[64 + f], hv, v);
  }
  su[r] = u; sv[r] = v;
  __syncthreads();
  expA[b * 1024 + r] = expf(lrelu02(su[r] + sv[r]));
  if (r < 512) expB[b * 512 + r] = expf(lrelu02(su[2 * r] + sv[2 * r + 1]));
}

// ---------------------------------------------------------------------------
// Kernel 3: denom[b,j] = sum_i exp(A0[b,i,j]) = 1024 + sum_i adj*(E-1);
//           then H[b,j,:] *= 1/denom  (in place -> becomes G)
// grid = B*(N/256) = 8 blocks, 256 threads; thread = column j (coalesced adj reads)
// ---------------------------------------------------------------------------
__global__ void k_denom_scale(const int* __restrict__ adj, const float* __restrict__ expA,
                              const float* __restrict__ expB, float* __restrict__ H) {
  const int b  = blockIdx.x >> 2;
  const int j  = ((blockIdx.x & 3) << 8) + threadIdx.x;
  const int jj = j & 511;
  const int sel = (j >= 512) ? 1 : 0;
  const float eBm1 = expB[b * 512 + jj] - 1.f;
  const int* col = adj + (size_t)b * 1024 * 1024 + j;

  float s = 0.f;
  for (int i = 0; i < 512; ++i) {
    const int av = col[(size_t)i * 1024];
    s += av ? (expA[b * 1024 + 2 * i + sel] - 1.f) : 0.f;
  }
  float cnt = 0.f;
  for (int i = 512; i < 1024; ++i) cnt += (float)col[(size_t)i * 1024];
  s = fmaf(cnt, eBm1, s);

  const float rd = 1.f / (1024.f + s);
  float* hrow = H + ((size_t)(b * 1024 + j)) * 64;
  #pragma unroll 8
  for (int f = 0; f < 64; ++f) hrow[f] *= rd;
}

// ---------------------------------------------------------------------------
// Kernel 4: out = elu( P @ G ), P[b,i,m] = adj ? E(i,m) : 1 synthesized on the fly.
//   i<512 : E = expA[b, 2i + (m>=512)]        i>=512 : E = expB[b, m%512]
// grid = B*(N/16) = 128 blocks, 128 threads = 4 waves; wave t owns cols [16t,16t+16)
// ---------------------------------------------------------------------------
__global__ void k_attn_gemm(const int* __restrict__ adj, const float* __restrict__ expA,
                            const float* __restrict__ expB, const float* __restrict__ G,
                            float* __restrict__ out) {
  const int b    = blockIdx.x >> 6;
  const int i0   = (blockIdx.x & 63) << 4;
  const int t    = threadIdx.x >> 5;
  const int lane = threadIdx.x & 31;
  const int mi = lane & 15, hh = lane >> 4;

  const int  ri    = i0 + mi;              // attention row this lane feeds into A-frag
  const bool caseA = (i0 < 512);
  const float e0 = caseA ? expA[b * 1024 + 2 * ri]     : 0.f;
  const float e1 = caseA ? expA[b * 1024 + 2 * ri + 1] : 0.f;
  const int*   arow = adj + ((size_t)(b * 1024 + ri)) * 1024;
  const float* eBb  = expB + b * 512;

  v8f acc = {};
  #pragma unroll 2
  for (int m0 = 0; m0 < 1024; m0 += 4) {
    const int mk = m0 + 2 * hh;            // even, so mk and mk+1 share the m<512 side
    const int av0 = arow[mk];
    const int av1 = arow[mk + 1];
    float E0, E1;
    if (caseA) { const float eA = (mk < 512) ? e0 : e1; E0 = eA; E1 = eA; }
    else       { E0 = eBb[mk & 511]; E1 = eBb[(mk + 1) & 511]; }
    v2f pa; pa[0] = av0 ? E0 : 1.f; pa[1] = av1 ? E1 : 1.f;

    v2f bb;
    bb[0] = G[((size_t)(b * 1024 + mk)) * 64 + t * 16 + mi];
    bb[1] = G[((size_t)(b * 1024 + mk + 1)) * 64 + t * 16 + mi];
    acc = WMMA_F32_4(pa, bb, acc);
  }

  #pragma unroll
  for (int r = 0; r < 8; ++r) {
    const int row = i0 + r + 8 * hh;
    const float x = acc[r];
    out[((size_t)(b * 1024 + row)) * 64 + t * 16 + mi] = (x > 0.f) ? x : expm1f(x);
  }
}

// ---------------------------------------------------------------------------
extern "C" void kernel_launch(void* const* d_in, const int* in_sizes, int n_in,
                              void* d_out, int out_size, void* d_ws, size_t ws_size,
                              hipStream_t stream) {
  const float* X    = (const float*)d_in[0];   // [2,1024,128] f32
  const int*   adj  = (const int*)  d_in[1];   // [2,1024,1024] i32
  const float* W    = (const float*)d_in[2];   // [128,64] f32
  const float* avec = (const float*)d_in[3];   // [128] f32
  float*       out  = (float*)d_out;           // [2,1024,64] f32

  char* ws = (char*)d_ws;
  float* H    = (float*)ws;                                   // 2*1024*64 f32 (512 KB)
  float* expA = (float*)(ws + (size_t)2 * 1024 * 64 * 4);     // 2*1024 f32
  float* expB = expA + 2 * 1024;                              // 2*512  f32

  k_gemm_h    <<<dim3(128), dim3(128), 0, stream>>>(X, W, H);
  k_scores    <<<dim3(2),   dim3(1024), 0, stream>>>(H, avec, expA, expB);
  k_denom_scale<<<dim3(8),  dim3(256), 0, stream>>>(adj, expA, expB, H);
  k_attn_gemm <<<dim3(128), dim3(128), 0, stream>>>(adj, expA, expB, H, out);
}